// SPDNetwork_1_53240414601662
// MI455X (gfx1250) — compile-verified
//
#include <hip/hip_runtime.h>
#include <hip/hip_bf16.h>
#include <math.h>

// SPDNet forward on gfx1250.
// One wave32 per SPD matrix; LDS-resident pipeline; all GEMMs via
// V_WMMA_F32_16X16X4_F32 (fp32 matrix path -- eigensolver chain needs fp32).

typedef __attribute__((ext_vector_type(2))) float v2f;
typedef __attribute__((ext_vector_type(8))) float v8f;

#define WPB 4              // waves (== matrices) per block
#define LD  33             // padded LDS row stride (fp32) to dodge bank conflicts
#define EPS_RE  1e-4f
#define EPS_LOG 1e-10f

// ---------------- WMMA f32 16x16x4 helpers (ISA 7.12.2 layouts) ----------------

__device__ inline v8f wmma4(v2f a, v2f b, v8f c) {
  // (neg_a, A, neg_b, B, c_mod, C, reuse_a, reuse_b)
  return __builtin_amdgcn_wmma_f32_16x16x4_f32(false, a, false, b, (short)0, c,
                                               false, false);
}

// A-frag 16x4: lanes 0-15 -> K={0,1}, lanes 16-31 -> K={2,3}; M = lane&15.
__device__ inline v2f ld_a(const float* M, int ld, int mrow, int kcol, int lane) {
  int h = lane >> 4, l = lane & 15;
  const float* p = M + (mrow + l) * ld + kcol + 2 * h;
  v2f a; a.x = p[0]; a.y = p[1];
  return a;
}
// A = srcT (A[m][k] = src[k][m])
__device__ inline v2f ld_a_t(const float* M, int ld, int mrow, int kcol, int lane) {
  int h = lane >> 4, l = lane & 15;
  v2f a;
  a.x = M[(kcol + 2 * h + 0) * ld + mrow + l];
  a.y = M[(kcol + 2 * h + 1) * ld + mrow + l];
  return a;
}
// B-frag 4x16: N = lane&15; VGPR0 rows {0,2}, VGPR1 rows {1,3} split by lane half.
__device__ inline v2f ld_b(const float* M, int ld, int krow, int ncol, int lane) {
  int h = lane >> 4, l = lane & 15;
  v2f b;
  b.x = M[(krow + 2 * h + 0) * ld + ncol + l];
  b.y = M[(krow + 2 * h + 1) * ld + ncol + l];
  return b;
}
// B = srcT (B[k][n] = src[n][k])
__device__ inline v2f ld_b_t(const float* M, int ld, int krow, int ncol, int lane) {
  int h = lane >> 4, l = lane & 15;
  v2f b;
  b.x = M[(ncol + l) * ld + krow + 2 * h + 0];
  b.y = M[(ncol + l) * ld + krow + 2 * h + 1];
  return b;
}
// D 16x16: VGPR r -> M = r (lanes 0-15) / r+8 (lanes 16-31); N = lane&15.
__device__ inline void st_d(float* M, int ld, int mrow, int ncol, int lane, v8f c) {
  int h = lane >> 4, l = lane & 15;
#pragma unroll
  for (int r = 0; r < 8; ++r)
    M[(mrow + r + 8 * h) * ld + ncol + l] = c[r];
}

// ---------------- parallel cyclic Jacobi (tournament schedule) ----------------

__device__ inline void jpair(int n, int r, int k, int* p, int* q) {
  // circle method: slot0 fixed = 0, slots 1..n-1 rotate by round r
  int a = (k == 0) ? 0 : (1 + (k - 1 + r) % (n - 1));
  int b = 1 + (n - 2 - k + r) % (n - 1);
  *p = a < b ? a : b;
  *q = a < b ? b : a;
}

// Symmetric Jacobi eigensolver on n x n LDS matrix (row stride ld).
// One wave per matrix; all waves in the block call with identical trip counts.
__device__ __attribute__((noinline))
void jacobi_sym(float* A, float* V, float* cs, int n, int lane, int sweeps) {
  for (int idx = lane; idx < n * n; idx += 32) {
    int i = idx / n, j = idx - i * n;
    V[i * LD + j] = (i == j) ? 1.0f : 0.0f;
  }
  __syncthreads();
  const int np = n >> 1;
#pragma unroll 1
  for (int s = 0; s < sweeps; ++s) {
#pragma unroll 1
    for (int r = 0; r < n - 1; ++r) {
      if (lane < np) {
        int p, q; jpair(n, r, lane, &p, &q);
        float app = A[p * LD + p], aqq = A[q * LD + q], apq = A[p * LD + q];
        float c = 1.0f, sn = 0.0f;
        if (fabsf(apq) > 1e-20f) {
          float tau = (aqq - app) / (2.0f * apq);
          float t = copysignf(1.0f, tau) / (fabsf(tau) + sqrtf(1.0f + tau * tau));
          c = rsqrtf(1.0f + t * t);
          sn = t * c;
        }
        cs[2 * lane + 0] = c;
        cs[2 * lane + 1] = sn;
      }
      __syncthreads();
      if (lane < n) {            // column update of A and V (right-multiply by J)
        int i = lane;
#pragma unroll 1
        for (int k = 0; k < np; ++k) {
          int p, q; jpair(n, r, k, &p, &q);
          float c = cs[2 * k], sn = cs[2 * k + 1];
          float t1 = A[i * LD + p], t2 = A[i * LD + q];
          A[i * LD + p] = c * t1 - sn * t2;
          A[i * LD + q] = sn * t1 + c * t2;
          float v1 = V[i * LD + p], v2 = V[i * LD + q];
          V[i * LD + p] = c * v1 - sn * v2;
          V[i * LD + q] = sn * v1 + c * v2;
        }
      }
      __syncthreads();
      if (lane < n) {            // row update of A (left-multiply by J^T)
        int j = lane;
#pragma unroll 1
        for (int k = 0; k < np; ++k) {
          int p, q; jpair(n, r, k, &p, &q);
          float c = cs[2 * k], sn = cs[2 * k + 1];
          float t1 = A[p * LD + j], t2 = A[q * LD + j];
          A[p * LD + j] = c * t1 - sn * t2;
          A[q * LD + j] = sn * t1 + c * t2;
        }
      }
      __syncthreads();
    }
  }
}

// ---------------- unrolled 4x4 register Jacobi for LogEig ----------------

template <int P, int Q>
__device__ inline void rot4(float (&A)[4][4], float (&V)[4][4]) {
  float app = A[P][P], aqq = A[Q][Q], apq = A[P][Q];
  float c = 1.0f, s = 0.0f;
  if (fabsf(apq) > 1e-20f) {
    float tau = (aqq - app) / (2.0f * apq);
    float t = copysignf(1.0f, tau) / (fabsf(tau) + sqrtf(1.0f + tau * tau));
    c = rsqrtf(1.0f + t * t);
    s = t * c;
  }
#pragma unroll
  for (int i = 0; i < 4; ++i) {
    float t1 = A[i][P], t2 = A[i][Q];
    A[i][P] = c * t1 - s * t2; A[i][Q] = s * t1 + c * t2;
  }
#pragma unroll
  for (int j = 0; j < 4; ++j) {
    float t1 = A[P][j], t2 = A[Q][j];
    A[P][j] = c * t1 - s * t2; A[Q][j] = s * t1 + c * t2;
  }
#pragma unroll
  for (int i = 0; i < 4; ++i) {
    float t1 = V[i][P], t2 = V[i][Q];
    V[i][P] = c * t1 - s * t2; V[i][Q] = s * t1 + c * t2;
  }
}

// ---------------- main kernel ----------------

__global__ __launch_bounds__(32 * WPB)
void spdnet_fwd(const float* __restrict__ x, const float* __restrict__ w1,
                const float* __restrict__ w2, const float* __restrict__ w3,
                const float* __restrict__ fcw, float* __restrict__ out, int Btot) {
  __shared__ float sA[WPB][32][LD];
  __shared__ float sV[WPB][32][LD];
  __shared__ float sT[WPB][32][LD];
  __shared__ float sW1[32][LD];
  __shared__ float sW2[32][17];
  __shared__ float sW3[16][5];
  __shared__ float sCS[WPB][32];

  const int tid = threadIdx.x;
  const int lane = tid & 31;
  const int wv = tid >> 5;
  int m = blockIdx.x * WPB + wv;
  if (m >= Btot) m = Btot - 1;   // keep barriers uniform on ragged tail

  // stage weights into LDS once per block
  for (int idx = tid; idx < 32 * 32; idx += blockDim.x)
    sW1[idx >> 5][idx & 31] = w1[idx];
  for (int idx = tid; idx < 32 * 16; idx += blockDim.x)
    sW2[idx >> 4][idx & 15] = w2[idx];
  for (int idx = tid; idx < 16 * 4; idx += blockDim.x)
    sW3[idx >> 2][idx & 3] = w3[idx];

  float* A = &sA[wv][0][0];
  float* V = &sV[wv][0][0];
  float* T = &sT[wv][0][0];
  float* CS = &sCS[wv][0];

  // load x[m] (coalesced: lane = column)
  const float* xm = x + (size_t)m * 1024;
#pragma unroll
  for (int r = 0; r < 32; ++r)
    A[r * LD + lane] = xm[r * 32 + lane];
  __syncthreads();

  // ---- stage 1: Y = W1^T x W1 (32x32) ----
#pragma unroll
  for (int mi = 0; mi < 2; ++mi)
#pragma unroll
    for (int ni = 0; ni < 2; ++ni) {
      v8f c = {};
#pragma unroll
      for (int kk = 0; kk < 8; ++kk)
        c = wmma4(ld_a(A, LD, mi * 16, kk * 4, lane),
                  ld_b(&sW1[0][0], LD, kk * 4, ni * 16, lane), c);
      st_d(T, LD, mi * 16, ni * 16, lane, c);          // T = x @ W1
    }
  __syncthreads();
#pragma unroll
  for (int mi = 0; mi < 2; ++mi)
#pragma unroll
    for (int ni = 0; ni < 2; ++ni) {
      v8f c = {};
#pragma unroll
      for (int kk = 0; kk < 8; ++kk)
        c = wmma4(ld_a_t(&sW1[0][0], LD, mi * 16, kk * 4, lane),
                  ld_b(T, LD, kk * 4, ni * 16, lane), c);
      st_d(A, LD, mi * 16, ni * 16, lane, c);          // A = W1^T @ T
    }
  __syncthreads();

  // ---- ReEig 32x32 ----
  jacobi_sym(A, V, CS, 32, lane, 10);
  {
    float wk = fmaxf(A[lane * LD + lane], EPS_RE);     // lane = eigen index k
#pragma unroll
    for (int i = 0; i < 32; ++i)
      T[i * LD + lane] = V[i * LD + lane] * wk;        // T = V * diag(w+)
  }
  __syncthreads();
#pragma unroll
  for (int mi = 0; mi < 2; ++mi)
#pragma unroll
    for (int ni = 0; ni < 2; ++ni) {
      v8f c = {};
#pragma unroll
      for (int kk = 0; kk < 8; ++kk)
        c = wmma4(ld_a(T, LD, mi * 16, kk * 4, lane),
                  ld_b_t(V, LD, kk * 4, ni * 16, lane), c);
      st_d(A, LD, mi * 16, ni * 16, lane, c);          // A = X1 = T @ V^T
    }
  __syncthreads();

  // ---- stage 2: Z = W2^T X1 W2 (16x16) ----
#pragma unroll
  for (int mi = 0; mi < 2; ++mi) {
    v8f c = {};
#pragma unroll
    for (int kk = 0; kk < 8; ++kk)
      c = wmma4(ld_a(A, LD, mi * 16, kk * 4, lane),
                ld_b(&sW2[0][0], 17, kk * 4, 0, lane), c);
    st_d(T, LD, mi * 16, 0, lane, c);                  // T(32x16) = X1 @ W2
  }
  __syncthreads();
  {
    v8f c = {};
#pragma unroll
    for (int kk = 0; kk < 8; ++kk)
      c = wmma4(ld_a_t(&sW2[0][0], 17, 0, kk * 4, lane),
                ld_b(T, LD, kk * 4, 0, lane), c);
    st_d(A, LD, 0, 0, lane, c);                        // A(16x16) = W2^T @ T
  }
  __syncthreads();

  // ---- ReEig 16x16 ----
  jacobi_sym(A, V, CS, 16, lane, 8);
  if (lane < 16) {
    float wk = fmaxf(A[lane * LD + lane], EPS_RE);
#pragma unroll
    for (int i = 0; i < 16; ++i)
      T[i * LD + lane] = V[i * LD + lane] * wk;
  }
  __syncthreads();
  {
    v8f c = {};
#pragma unroll
    for (int kk = 0; kk < 4; ++kk)
      c = wmma4(ld_a(T, LD, 0, kk * 4, lane),
                ld_b_t(V, LD, kk * 4, 0, lane), c);
    st_d(A, LD, 0, 0, lane, c);                        // A = X2 = T @ V^T
  }
  __syncthreads();

  // ---- stage 3: M3 = W3^T X2 W3 (4x4), lanes 0..15 each one element ----
  if (lane < 16) {
    int i3 = lane >> 2, j3 = lane & 3;
    float acc = 0.0f;
#pragma unroll 1
    for (int l2 = 0; l2 < 16; ++l2) {
      float u = 0.0f;
#pragma unroll
      for (int k2 = 0; k2 < 16; ++k2)
        u += sW3[k2][i3] * A[k2 * LD + l2];
      acc += u * sW3[l2][j3];
    }
    T[i3 * LD + j3] = acc;
  }
  __syncthreads();

  // ---- LogEig 4x4 + FC + log_softmax (lane 0) ----
  if (lane == 0) {
    float a4[4][4], v4[4][4];
#pragma unroll
    for (int i = 0; i < 4; ++i)
#pragma unroll
      for (int j = 0; j < 4; ++j) {
        a4[i][j] = T[i * LD + j];
        v4[i][j] = (i == j) ? 1.0f : 0.0f;
      }
#pragma unroll
    for (int s = 0; s < 8; ++s) {
      rot4<0, 1>(a4, v4); rot4<0, 2>(a4, v4); rot4<0, 3>(a4, v4);
      rot4<1, 2>(a4, v4); rot4<1, 3>(a4, v4); rot4<2, 3>(a4, v4);
    }
    float lw[4];
#pragma unroll
    for (int k = 0; k < 4; ++k)
      lw[k] = logf(fmaxf(a4[k][k], EPS_LOG));
    float feat[16];
#pragma unroll
    for (int i = 0; i < 4; ++i)
#pragma unroll
      for (int j = 0; j < 4; ++j) {
        float acc = 0.0f;
#pragma unroll
        for (int k = 0; k < 4; ++k)
          acc += v4[i][k] * lw[k] * v4[j][k];
        feat[i * 4 + j] = acc;
      }
    float* featOut = out + (size_t)Btot * 2 + (size_t)m * 16;
#pragma unroll
    for (int t = 0; t < 16; ++t)
      featOut[t] = feat[t];
    float l0 = 0.0f, l1 = 0.0f;
#pragma unroll
    for (int t = 0; t < 16; ++t) {
      l0 += feat[t] * fcw[t * 2 + 0];
      l1 += feat[t] * fcw[t * 2 + 1];
    }
    float mx = fmaxf(l0, l1);
    float lse = mx + logf(expf(l0 - mx) + expf(l1 - mx));
    out[(size_t)m * 2 + 0] = l0 - lse;
    out[(size_t)m * 2 + 1] = l1 - lse;
  }
}

extern "C" void kernel_launch(void* const* d_in, const int* in_sizes, int n_in,
                              void* d_out, int out_size, void* d_ws, size_t ws_size,
                              hipStream_t stream) {
  const float* x   = (const float*)d_in[0];
  const float* w1  = (const float*)d_in[1];
  const float* w2  = (const float*)d_in[2];
  const float* w3  = (const float*)d_in[3];
  const float* fcw = (const float*)d_in[4];
  float* out = (float*)d_out;
  int Btot = in_sizes[0] / 1024;     // B*32*32 elements -> B
  int blocks = (Btot + WPB - 1) / WPB;
  spdnet_fwd<<<blocks, 32 * WPB, 0, stream>>>(x, w1, w2, w3, fcw, out, Btot);
}